// AttentionDownsampler_20478404067693
// MI455X (gfx1250) — compile-verified
//
#include <hip/hip_runtime.h>
#include <math.h>

typedef __attribute__((ext_vector_type(2))) float v2f;
typedef __attribute__((ext_vector_type(8))) float v8f;

#define DIM      384
#define KK       7
#define PP       49          // KK*KK
#define FINAL    16
#define HH       112
#define PSTRIDE  7
#define PADC     388         // padded LDS channel stride (mod 64 = 4 -> conflict-free wmma A loads)
#define NTHREADS 128

__global__ __launch_bounds__(NTHREADS)
void attn_downsample_kernel(const float* __restrict__ hr,
                            const float* __restrict__ du,
                            const float* __restrict__ aw,
                            const float* __restrict__ ab,
                            const float* __restrict__ wmat,
                            const float* __restrict__ bmat,
                            float* __restrict__ out)
{
    __shared__ float sPatch[PP][PADC];   // 49 x 388 f32 = 76048 B
    __shared__ float sAw[DIM];
    __shared__ float sW[64];
    __shared__ float sB[64];
    __shared__ float sLogits[64];
    __shared__ float sAttn[64];

    const int t  = threadIdx.x;
    const int bx = blockIdx.x;
    const int ow = bx & 15;
    const int oh = (bx >> 4) & 15;
    const int bb = bx >> 8;

    // ---- stage small vectors (regular loads; tiny) ----
    for (int c = t; c < DIM; c += NTHREADS) sAw[c] = aw[c];
    if (t < 64) {
        sW[t] = (t < PP) ? wmat[t] : 0.0f;
        sB[t] = (t < PP) ? bmat[t] : 0.0f;
    }

    // ---- stage patch via async global->LDS gather (ASYNCcnt path) ----
    // hr[bb, c, 7*oh+dr, 7*ow+dc] -> sPatch[7*dr+dc][c]
    const float* hrb = hr + (long)bb * DIM * HH * HH;          // uniform -> SGPR pair
    const int    y0  = oh * PSTRIDE;
    const int    x0  = ow * PSTRIDE;
    const unsigned sbase = (unsigned)(uintptr_t)&sPatch[0][0]; // flat LDS aperture: low 32 bits = LDS addr
    for (int e = t; e < PP * DIM; e += NTHREADS) {             // 147 full iterations, EXEC all-1
        const int c  = e / PP;
        const int p  = e - c * PP;
        const int dr = p / KK;
        const int dc = p - dr * KK;
        const unsigned goff  = (unsigned)((((c * HH) + (y0 + dr)) * HH + (x0 + dc)) * 4);
        const unsigned laddr = sbase + (unsigned)((p * PADC + c) * 4);
        asm volatile("global_load_async_to_lds_b32 %0, %1, %2"
                     :: "v"(laddr), "v"(goff), "s"(hrb) : "memory");
    }
    asm volatile("s_wait_asynccnt 0" ::: "memory");
    __syncthreads();

    if (t < 32) {
        // ---- wave 0: logits p=0..47 via V_WMMA_F32_16X16X4_F32 ----
        // A tile: 16 patch rows x 4 channels (lane<16: K=c0,c0+1 ; lane>=16: K=c0+2,c0+3)
        // B tile: aw chunk broadcast into all 16 N columns -> every D column = logits
        const int lane = t;
        const int half = lane >> 4;   // 0 or 1
        const int l16  = lane & 15;
        v8f acc0 = {}; v8f acc1 = {}; v8f acc2 = {};
        for (int k = 0; k < DIM / 4; ++k) {
            const int c0 = 4 * k + 2 * half;
            v2f bv;
            bv.x = sAw[c0];
            bv.y = sAw[c0 + 1];
            v2f a0 = *(const v2f*)&sPatch[ 0 + l16][c0];
            v2f a1 = *(const v2f*)&sPatch[16 + l16][c0];
            v2f a2 = *(const v2f*)&sPatch[32 + l16][c0];
            acc0 = __builtin_amdgcn_wmma_f32_16x16x4_f32(false, a0, false, bv, (short)0, acc0, false, false);
            acc1 = __builtin_amdgcn_wmma_f32_16x16x4_f32(false, a1, false, bv, (short)0, acc1, false, false);
            acc2 = __builtin_amdgcn_wmma_f32_16x16x4_f32(false, a2, false, bv, (short)0, acc2, false, false);
        }
        // D layout: VGPR j, lane l -> M = j + 8*(l/16), N = l%16 (all N columns equal here)
        if (l16 == 0) {
            #pragma unroll
            for (int j = 0; j < 8; ++j) {
                sLogits[ 0 + 8 * half + j] = acc0[j];
                sLogits[16 + 8 * half + j] = acc1[j];
                sLogits[32 + 8 * half + j] = acc2[j];
            }
        }
    } else if (t < 64) {
        // ---- wave 1 (concurrent): logits p=48 via VALU dot + wave32 reduction ----
        const int lane = t - 32;
        float acc = 0.0f;
        for (int c = lane; c < DIM; c += 32) acc += sPatch[48][c] * sAw[c];
        #pragma unroll
        for (int off = 16; off > 0; off >>= 1) acc += __shfl_xor(acc, off, 32);
        if (lane == 0) sLogits[48] = acc;
    }
    __syncthreads();

    // ---- softmax over 49 (mask is a scalar per (b,oh,ow)) ----
    const float msk = (du[(bb * FINAL + oh) * FINAL + ow] > 0.2f) ? 1.0f : 0.0f;
    const float abv = ab[0];
    float mx = -3.0e38f;
    for (int p = 0; p < PP; ++p) {
        const float al = (sLogits[p] + abv) * msk * sW[p] + sB[p];
        mx = fmaxf(mx, al);
    }
    float sum = 0.0f;
    for (int p = 0; p < PP; ++p) {
        const float al = (sLogits[p] + abv) * msk * sW[p] + sB[p];
        sum += expf(al - mx);
    }
    if (t < PP) {
        const float al = (sLogits[t] + abv) * msk * sW[t] + sB[t];
        sAttn[t] = expf(al - mx) / sum;
    }
    __syncthreads();

    // ---- out[b, c, oh, ow] = sum_p patch[p][c] * attn[p] ----
    for (int c = t; c < DIM; c += NTHREADS) {
        float a = 0.0f;
        for (int p = 0; p < PP; ++p) a += sPatch[p][c] * sAttn[p];
        out[(((long)bb * DIM + c) * FINAL + oh) * FINAL + ow] = a;
    }
}

extern "C" void kernel_launch(void* const* d_in, const int* in_sizes, int n_in,
                              void* d_out, int out_size, void* d_ws, size_t ws_size,
                              hipStream_t stream) {
    // input order: hr_feats, guidance(unused), dropout_u, aw, ab, w, b
    const float* hr = (const float*)d_in[0];
    const float* du = (const float*)d_in[2];
    const float* aw = (const float*)d_in[3];
    const float* ab = (const float*)d_in[4];
    const float* w  = (const float*)d_in[5];
    const float* b  = (const float*)d_in[6];
    float* out = (float*)d_out;

    dim3 grid(8 * FINAL * FINAL);   // one block per (b, oh, ow)
    dim3 block(NTHREADS);
    hipLaunchKernelGGL(attn_downsample_kernel, grid, block, 0, stream,
                       hr, du, aw, ab, w, b, out);
}